// CNNTest_10299331576114
// MI455X (gfx1250) — compile-verified
//
#include <hip/hip_runtime.h>
#include <hip/hip_bf16.h>

typedef float v2f __attribute__((ext_vector_type(2)));
typedef float v8f __attribute__((ext_vector_type(8)));

// ---------------- Stage 1: h[v] = mean_n relu(k3conv_n(vp[nb1[v,:]])) ---------
// One wave32 per vertex, lane = neighbor index. Conv over the lane axis via
// lane shuffles (zero padding at the ends), then a 5-step shfl_xor reduction.
__global__ void k_h1(const float* __restrict__ vp, const int* __restrict__ nb1,
                     const float* __restrict__ wv1, const float* __restrict__ bv1,
                     float* __restrict__ h, int V) {
  int wave = (int)((blockIdx.x * blockDim.x + threadIdx.x) >> 5);
  int lane = threadIdx.x & 31;
  if (wave >= V) return;
  float g = vp[nb1[(size_t)wave * 32 + lane]];
  float gl = __shfl_up(g, 1, 32);   if (lane == 0)  gl = 0.f;
  float gr = __shfl_down(g, 1, 32); if (lane == 31) gr = 0.f;
  float w0 = wv1[0], w1 = wv1[1], w2 = wv1[2], b = bv1[0];
  float c = fmaxf(w0 * gl + w1 * g + w2 * gr + b, 0.f);
  #pragma unroll
  for (int off = 16; off > 0; off >>= 1) c += __shfl_xor(c, off, 32);
  if (lane == 0) h[wave] = c * (1.f / 32.f);
}

// ---------------- Stage 2: f1[v,o] = b1[o] + sum_k w1[o,k]*h[v-1+k] ----------
__global__ void k_f1(const float* __restrict__ h, const float* __restrict__ w1,
                     const float* __restrict__ b1, float* __restrict__ f1, int V) {
  int v = (int)((blockIdx.x * blockDim.x + threadIdx.x) >> 5);
  int o = threadIdx.x & 31;
  if (v >= V) return;
  float hm = (v > 0)     ? h[v - 1] : 0.f;
  float h0 = h[v];
  float hp = (v + 1 < V) ? h[v + 1] : 0.f;
  f1[(size_t)v * 32 + o] = b1[o] + w1[o * 3 + 0] * hm + w1[o * 3 + 1] * h0
                                  + w1[o * 3 + 2] * hp;
}

// ---- Stage 3: h2[v,c] = mean_n relu(k3conv_n(f1[nb2[v,n],c])) ---------------
// One wave per vertex, lane = channel -> each neighbor row is one coalesced
// 128B load (f1 table is 12.8MB, L2-resident). Rolling prev/cur/nxt window.
__global__ void k_h2(const float* __restrict__ f1, const int* __restrict__ nb2,
                     const float* __restrict__ wv2, const float* __restrict__ bv2,
                     float* __restrict__ h2, int V) {
  int v = (int)((blockIdx.x * blockDim.x + threadIdx.x) >> 5);
  int c = threadIdx.x & 31;
  if (v >= V) return;
  const int* nbr = nb2 + (size_t)v * 32;
  float w0 = wv2[0], w1 = wv2[1], w2 = wv2[2], b = bv2[0];
  float prev = 0.f;
  float cur = f1[(size_t)nbr[0] * 32 + c];
  float acc = 0.f;
  #pragma unroll
  for (int n = 0; n < 32; ++n) {
    float nxt = (n < 31) ? f1[(size_t)nbr[n + 1] * 32 + c] : 0.f;
    acc += fmaxf(w0 * prev + w1 * cur + w2 * nxt + b, 0.f);
    prev = cur; cur = nxt;
  }
  h2[(size_t)v * 32 + c] = acc * (1.f / 32.f);
}

// ---- Stage 4: f2[v,o] = b2[o] + sum_{k,c} h2[v-1+k,c] * w2[o,c,k] -----------
// w2 (24KB) staged in LDS; 4 vertices x 64 outputs per 256-thread block.
__global__ void k_f2(const float* __restrict__ h2, const float* __restrict__ w2,
                     const float* __restrict__ b2, float* __restrict__ f2, int V) {
  __shared__ float lw2[64 * 96];
  for (int i = threadIdx.x; i < 64 * 96; i += blockDim.x) lw2[i] = w2[i];
  __syncthreads();
  int o  = threadIdx.x & 63;
  int vi = threadIdx.x >> 6;
  int v  = blockIdx.x * 4 + vi;
  if (v >= V) return;
  float acc = b2[o];
  #pragma unroll
  for (int k = 0; k < 3; ++k) {
    int vv = v - 1 + k;
    if ((unsigned)vv < (unsigned)V) {
      const float* hr = h2 + (size_t)vv * 32;
      #pragma unroll
      for (int c = 0; c < 32; ++c) acc += hr[c] * lw2[o * 96 + c * 3 + k];
    }
  }
  f2[(size_t)v * 64 + o] = acc;
}

// ---- Stage 5: logits = f2 @ wfc^T + bfc ; softmax over 512 ------------------
// One wave per 16-vertex tile. Full-precision V_WMMA_F32_16X16X4_F32:
//   A (16x4 f32): lane L holds M=L&15, K = 2*(L>>4)+{0,1}  (8B loads from f2)
//   B (4x16 f32): lane L holds N=L&15, K = 2*(L>>4)+{0,1}  (8B loads from wfc row)
//   C/D (16x16):  c[r] = C[r + 8*(L>>4)][L&15], seeded with bias.
// 32 N-tiles x 16 K-steps = 512 WMMA per wave; logits staged in 32KB LDS so
// the softmax is fused (saves a 410MB logits round-trip -> we stay on the
// 205MB-output bandwidth roofline, ~9us at 23.3 TB/s).
__global__ void __launch_bounds__(32)
k_fc_softmax(const float* __restrict__ f2, const float* __restrict__ wfc,
             const float* __restrict__ bfc, float* __restrict__ out, int V) {
  __shared__ float lrow[16 * 512];  // 32 KB
  int nTiles = V >> 4;
  int tile = blockIdx.x;
  if (tile >= nTiles) return;           // wave-uniform: EXEC stays all-1s
  int lane = threadIdx.x;
  int nlow = lane & 15;
  int hi   = lane >> 4;

  int v0 = tile * 16;
  // Preload all 16 A fragments (K=64) once; reused across all 32 N-tiles.
  const float* fr = f2 + (size_t)(v0 + nlow) * 64 + hi * 2;
  v2f a[16];
  #pragma unroll
  for (int kk = 0; kk < 16; ++kk) a[kk] = *(const v2f*)(fr + kk * 4);

  for (int nt = 0; nt < 32; ++nt) {
    int n = nt * 16 + nlow;
    float bias = bfc[n];
    v8f acc;
    #pragma unroll
    for (int i = 0; i < 8; ++i) acc[i] = bias;
    const float* wr = wfc + (size_t)n * 64 + hi * 2;
    #pragma unroll
    for (int kk = 0; kk < 16; ++kk) {
      v2f bfrag = *(const v2f*)(wr + kk * 4);
      acc = __builtin_amdgcn_wmma_f32_16x16x4_f32(
          /*neg_a=*/false, a[kk], /*neg_b=*/false, bfrag,
          /*c_mod=*/(short)0, acc, /*reuse_a=*/false, /*reuse_b=*/false);
    }
    #pragma unroll
    for (int r = 0; r < 8; ++r)
      lrow[(r + 8 * hi) * 512 + nt * 16 + nlow] = acc[r];
  }
  __syncthreads();

  // Row softmax: wave-cooperative max / sum-exp, fully coalesced writes.
  for (int r = 0; r < 16; ++r) {
    float m = -3.402823466e38f;
    #pragma unroll
    for (int i = 0; i < 16; ++i) m = fmaxf(m, lrow[r * 512 + i * 32 + lane]);
    #pragma unroll
    for (int off = 16; off > 0; off >>= 1) m = fmaxf(m, __shfl_xor(m, off, 32));
    float s = 0.f;
    #pragma unroll
    for (int i = 0; i < 16; ++i) {
      float e = __expf(lrow[r * 512 + i * 32 + lane] - m);
      lrow[r * 512 + i * 32 + lane] = e;
      s += e;
    }
    #pragma unroll
    for (int off = 16; off > 0; off >>= 1) s += __shfl_xor(s, off, 32);
    float inv = 1.f / s;
    float* orow = out + (size_t)(v0 + r) * 512;
    #pragma unroll
    for (int i = 0; i < 16; ++i)
      orow[i * 32 + lane] = lrow[r * 512 + i * 32 + lane] * inv;
  }
}

extern "C" void kernel_launch(void* const* d_in, const int* in_sizes, int n_in,
                              void* d_out, int out_size, void* d_ws, size_t ws_size,
                              hipStream_t stream) {
  const float* vp  = (const float*)d_in[0];
  const int*   nb1 = (const int*)  d_in[1];
  const int*   nb2 = (const int*)  d_in[2];
  const float* wv1 = (const float*)d_in[3];
  const float* bv1 = (const float*)d_in[4];
  const float* w1  = (const float*)d_in[5];
  const float* b1  = (const float*)d_in[6];
  const float* wv2 = (const float*)d_in[7];
  const float* bv2 = (const float*)d_in[8];
  const float* w2  = (const float*)d_in[9];
  const float* b2  = (const float*)d_in[10];
  const float* wfc = (const float*)d_in[11];
  const float* bfc = (const float*)d_in[12];
  float* out = (float*)d_out;
  const int V = in_sizes[0];

  // Workspace partition (all f32): h[V] | f1[V*32] | h2[V*32] | f2[V*64]
  float* h  = (float*)d_ws;
  float* f1 = h  + V;
  float* h2 = f1 + (size_t)V * 32;
  float* f2 = h2 + (size_t)V * 32;

  const int wavesPerBlock = 8;                       // 256 threads = 8 wave32
  const int nbW = (V + wavesPerBlock - 1) / wavesPerBlock;
  k_h1<<<nbW, 256, 0, stream>>>(vp, nb1, wv1, bv1, h, V);
  k_f1<<<nbW, 256, 0, stream>>>(h, w1, b1, f1, V);
  k_h2<<<nbW, 256, 0, stream>>>(f1, nb2, wv2, bv2, h2, V);
  k_f2<<<(V + 3) / 4, 256, 0, stream>>>(h2, w2, b2, f2, V);
  const int nTiles = (V + 15) / 16;                  // V=100000 -> 6250 tiles
  k_fc_softmax<<<nTiles, 32, 0, stream>>>(f2, wfc, bfc, out, V);
}